// MAF_76029511074009
// MI455X (gfx1250) — compile-verified
//
#include <hip/hip_runtime.h>
#include <hip/hip_bf16.h>

#define Nn_  50000
#define Ee_  150000
#define Dd_  128
#define Hh_  4
#define HD_  512
#define ED_  16

typedef float v2f __attribute__((ext_vector_type(2)));
typedef float v8f __attribute__((ext_vector_type(8)));
typedef unsigned int u32x4 __attribute__((ext_vector_type(4)));
typedef int i32x4 __attribute__((ext_vector_type(4)));
typedef int i32x8 __attribute__((ext_vector_type(8)));

__device__ __forceinline__ float waveSum(float v) {
#pragma unroll
  for (int off = 16; off > 0; off >>= 1) v += __shfl_xor(v, off, 32);
  return v;
}

// float atomic max via signed/unsigned integer ordering trick
__device__ __forceinline__ void atomicMaxF(float* addr, float val) {
  if (val >= 0.f) atomicMax((int*)addr, __float_as_int(val));
  else            atomicMin((unsigned int*)addr, __float_as_uint(val));
}

__global__ void fill_kernel(float* __restrict__ p, long long n, float v) {
  long long i = (long long)blockIdx.x * blockDim.x + threadIdx.x;
  long long stride = (long long)gridDim.x * blockDim.x;
  for (; i < n; i += stride) p[i] = v;
}

// ---------------------------------------------------------------------------
// WMMA f32 GEMM with TDM-staged weight tile:
//   C[M x Nout](ldc) = act( A[M x K](lda) @ W[K x Nout](ldw, +wcol0) + bias
//                           + (acc ? C_in : 0) )
// Block = 8 waves = 128 rows x 16 cols. The K x 16 weight tile is DMA'd once
// per block into LDS by the Tensor Data Mover (tensor_load_to_lds, 2D D#),
// then every wave reads B fragments from LDS. One V_WMMA_F32_16X16X4_F32 per
// k-step per wave. K multiple of 4 (<=384), M/Nout multiples of 16.
// ---------------------------------------------------------------------------
__global__ void gemm_wmma(const float* __restrict__ A, int lda,
                          const float* __restrict__ Wm, int ldw, int wcol0,
                          const float* __restrict__ bias,
                          float* __restrict__ C, int ldc,
                          int M, int K, int Nout, int accflag, int reluflag) {
  __shared__ float Bs[384 * 16];               // 24 KB: staged K x 16 weight tile
  const int wave = threadIdx.x >> 5;
  const int lane = threadIdx.x & 31;
  const int row0 = (blockIdx.y * 8 + wave) * 16;
  const int col0 = blockIdx.x * 16;

  if (wave == 0) {
    // Tensor DMA descriptor (D#): 2D tile, K rows x 16 cols of f32,
    // global row stride = ldw elements, contiguous into LDS at Bs.
    unsigned long long ga =
        (unsigned long long)(const void*)(Wm + (size_t)wcol0 + col0);
    unsigned int ldsa = (unsigned int)(unsigned long long)(const void*)&Bs[0];
    u32x4 g0;
    g0.x = 1u;                                            // count=1, user mode
    g0.y = ldsa;                                          // lds_addr
    g0.z = (unsigned int)(ga & 0xffffffffull);            // global_addr[31:0]
    g0.w = (unsigned int)((ga >> 32) & 0x1ffffffull)      // global_addr[56:32]
         | (2u << 30);                                    // type = 2 ("image")
    i32x8 g1;
    g1[0] = (int)(2u << 16);                              // data_size = 4 bytes
    g1[1] = (int)(16u << 16);                             // tensor_dim0 = 16 (lo)
    g1[2] = (int)(((unsigned int)K & 0xffffu) << 16);     // tdim0 hi=0 | tensor_dim1 lo
    g1[3] = (int)(((unsigned int)K >> 16) | (16u << 16)); // tdim1 hi | tile_dim0 = 16
    g1[4] = (int)((unsigned int)K & 0xffffu);             // tile_dim1 = K, tile_dim2 = 0
    g1[5] = ldw;                                          // tensor_dim0_stride (lo)
    g1[6] = 0;
    g1[7] = 0;                                            // dim1 stride unused (2D)
    i32x4 gz  = {0, 0, 0, 0};                             // groups 2/3 unused
    i32x8 gz8 = {0, 0, 0, 0, 0, 0, 0, 0};                 // extra group (clang-23 form)
    __builtin_amdgcn_tensor_load_to_lds(g0, g1, gz, gz, gz8, 0);
    __builtin_amdgcn_s_wait_tensorcnt(0);
  }
  __syncthreads();                                        // publish LDS tile

  if (row0 < M) {                                         // wave-uniform guard
    const int r = lane & 15;                              // A-row / B,C-col in tile
    const int khalf = lane >> 4;                          // K pairing per ISA layout
    v8f c = {};
    const float* Arow = A + (size_t)(row0 + r) * lda;
    for (int k0 = 0; k0 < K; k0 += 4) {
      const int ka = k0 + khalf * 2;
      v2f a, b;
      a.x = Arow[ka];
      a.y = Arow[ka + 1];
      b.x = Bs[ka * 16 + r];
      b.y = Bs[(ka + 1) * 16 + r];
      c = __builtin_amdgcn_wmma_f32_16x16x4_f32(false, a, false, b,
                                                (short)0, c, false, false);
    }
    const float bv = bias ? bias[col0 + r] : 0.f;
#pragma unroll
    for (int i = 0; i < 8; ++i) {
      size_t idx = (size_t)(row0 + khalf * 8 + i) * ldc + col0 + r;
      float v = c[i] + bv;
      if (accflag) v += C[idx];
      if (reluflag) v = fmaxf(v, 0.f);
      C[idx] = v;
    }
  }
}

// ---------------------------------------------------------------------------
// SAGE: scatter-sum neighbors + degree, then mean in place
// ---------------------------------------------------------------------------
__global__ void sage_scatter(const int* __restrict__ src, const int* __restrict__ dst,
                             const float* __restrict__ x,
                             float* __restrict__ agg, float* __restrict__ cnt) {
  int e = blockIdx.x;
  if (e >= Ee_) return;
  int t = threadIdx.x;                          // 128 threads = one feature row
  int s = src[e], d = dst[e];
  atomicAdd(&agg[(size_t)d * Dd_ + t], x[(size_t)s * Dd_ + t]);
  if (t == 0) atomicAdd(&cnt[d], 1.0f);
}

__global__ void mean_kernel(float* __restrict__ agg, const float* __restrict__ cnt) {
  size_t i = (size_t)blockIdx.x * blockDim.x + threadIdx.x;
  if (i >= (size_t)Nn_ * Dd_) return;
  agg[i] /= fmaxf(cnt[i >> 7], 1.0f);
}

// ---------------------------------------------------------------------------
// TransformerConv helpers (per head h)
// ---------------------------------------------------------------------------
// P[n,j] = sum_d q_h[n,d] * te_W[j, h*128+d]
__global__ void trans_P(const float* __restrict__ q_h, const float* __restrict__ te_W,
                        int h, float* __restrict__ P) {
  int wave = threadIdx.x >> 5, lane = threadIdx.x & 31;
  int n = blockIdx.x * 8 + wave;
  if (n >= Nn_) return;
  float4 q4 = *(const float4*)(q_h + (size_t)n * Dd_ + lane * 4);
  for (int j = 0; j < ED_; ++j) {
    float4 t4 = *(const float4*)(te_W + (size_t)j * HD_ + h * Dd_ + lane * 4);
    float s = q4.x * t4.x + q4.y * t4.y + q4.z * t4.z + q4.w * t4.w;
    s = waveSum(s);
    if (lane == 0) P[(size_t)n * ED_ + j] = s;
  }
}

// logits[e] = (q[dst]·k[src] + sum_j ea[e,j]*P[dst,j]) / sqrt(128); atomic-max per dst
__global__ void trans_logits(const int* __restrict__ src, const int* __restrict__ dst,
                             const float* __restrict__ q_h, const float* __restrict__ k_h,
                             const float* __restrict__ v_h,
                             const float* __restrict__ ea, const float* __restrict__ P,
                             float* __restrict__ tlog, float* __restrict__ tmax) {
  int wave = threadIdx.x >> 5, lane = threadIdx.x & 31;
  int e = blockIdx.x * 8 + wave;
  if (e >= Ee_) return;
  int s = src[e], t = dst[e];
  __builtin_prefetch(v_h + (size_t)s * Dd_ + lane * 4, 0, 1);  // warm L2 for accum pass
  float4 qv = *(const float4*)(q_h + (size_t)t * Dd_ + lane * 4);
  float4 kv = *(const float4*)(k_h + (size_t)s * Dd_ + lane * 4);
  float p = qv.x * kv.x + qv.y * kv.y + qv.z * kv.z + qv.w * kv.w;
  if (lane < ED_) p += ea[(size_t)e * ED_ + lane] * P[(size_t)t * ED_ + lane];
  float tot = waveSum(p) * 0.08838834764831845f;   // 1/sqrt(128)
  if (lane == 0) { tlog[e] = tot; atomicMaxF(&tmax[t], tot); }
}

// w = exp(l - max); scatter sums + v-numerator + edge-attr weighted sums
__global__ void trans_accum(const int* __restrict__ src, const int* __restrict__ dst,
                            const float* __restrict__ v_h, const float* __restrict__ ea,
                            const float* __restrict__ tlog, const float* __restrict__ tmax,
                            float* __restrict__ tsum, float* __restrict__ VN,
                            float* __restrict__ WA) {
  int wave = threadIdx.x >> 5, lane = threadIdx.x & 31;
  int e = blockIdx.x * 8 + wave;
  if (e >= Ee_) return;
  int s = src[e], t = dst[e];
  float w = __expf(tlog[e] - tmax[t]);
  if (lane == 0) atomicAdd(&tsum[t], w);
  float4 vv = *(const float4*)(v_h + (size_t)s * Dd_ + lane * 4);
  float* vn = VN + (size_t)t * Dd_ + lane * 4;
  atomicAdd(vn + 0, w * vv.x);
  atomicAdd(vn + 1, w * vv.y);
  atomicAdd(vn + 2, w * vv.z);
  atomicAdd(vn + 3, w * vv.w);
  if (lane < ED_) atomicAdd(&WA[(size_t)t * ED_ + lane], w * ea[(size_t)e * ED_ + lane]);
}

// x_attn_acc[n,d] += (VN + WA @ te_W_h) / (sum + 1e-16) / H
__global__ void trans_node(const float* __restrict__ VN, const float* __restrict__ WA,
                           const float* __restrict__ tsum, const float* __restrict__ te_W,
                           int h, float* __restrict__ xattn) {
  int n = blockIdx.x, d = threadIdx.x;
  float corr = 0.f;
#pragma unroll
  for (int j = 0; j < ED_; ++j)
    corr += WA[(size_t)n * ED_ + j] * te_W[(size_t)j * HD_ + h * Dd_ + d];
  float num = VN[(size_t)n * Dd_ + d] + corr;
  xattn[(size_t)n * Dd_ + d] += num / (tsum[n] + 1e-16f) * 0.25f;
}

__global__ void add_relu(const float* __restrict__ a, const float* __restrict__ b,
                         float* __restrict__ o, long long n) {
  long long i = (long long)blockIdx.x * blockDim.x + threadIdx.x;
  if (i < n) o[i] = fmaxf(a[i] + b[i], 0.f);
}

__global__ void bias_relu(const float* __restrict__ a, const float* __restrict__ bias,
                          float* __restrict__ o, long long n) {
  long long i = (long long)blockIdx.x * blockDim.x + threadIdx.x;
  if (i < n) o[i] = fmaxf(a[i] + bias[i & (Dd_ - 1)], 0.f);
}

// ---------------------------------------------------------------------------
// GAT helpers (per head h); index space E+N covers self-loops
// ---------------------------------------------------------------------------
__global__ void gat_ad(const float* __restrict__ g_h, const float* __restrict__ att_src,
                       const float* __restrict__ att_dst, int h,
                       float* __restrict__ a_s, float* __restrict__ a_d) {
  int wave = threadIdx.x >> 5, lane = threadIdx.x & 31;
  int n = blockIdx.x * 8 + wave;
  if (n >= Nn_) return;
  float4 gv = *(const float4*)(g_h + (size_t)n * Dd_ + lane * 4);
  float4 as4 = *(const float4*)(att_src + (size_t)h * Dd_ + lane * 4);
  float4 ad4 = *(const float4*)(att_dst + (size_t)h * Dd_ + lane * 4);
  float ps = gv.x * as4.x + gv.y * as4.y + gv.z * as4.z + gv.w * as4.w;
  float pd = gv.x * ad4.x + gv.y * ad4.y + gv.z * ad4.z + gv.w * ad4.w;
  ps = waveSum(ps);
  pd = waveSum(pd);
  if (lane == 0) { a_s[n] = ps; a_d[n] = pd; }
}

__global__ void gat_logits(const int* __restrict__ src, const int* __restrict__ dst,
                           const float* __restrict__ a_s, const float* __restrict__ a_d,
                           float* __restrict__ glog, float* __restrict__ gmax) {
  int i = blockIdx.x * blockDim.x + threadIdx.x;
  if (i >= Ee_ + Nn_) return;
  int s, t;
  if (i < Ee_) { s = src[i]; t = dst[i]; } else { s = t = i - Ee_; }
  float lg = a_s[s] + a_d[t];
  lg = lg > 0.f ? lg : 0.2f * lg;              // leaky_relu(0.2)
  glog[i] = lg;
  atomicMaxF(&gmax[t], lg);
}

__global__ void gat_accum(const int* __restrict__ src, const int* __restrict__ dst,
                          const float* __restrict__ g_h, const float* __restrict__ glog,
                          const float* __restrict__ gmax, float* __restrict__ gsum,
                          float* __restrict__ GN) {
  int wave = threadIdx.x >> 5, lane = threadIdx.x & 31;
  int i = blockIdx.x * 8 + wave;
  if (i >= Ee_ + Nn_) return;
  int s, t;
  if (i < Ee_) { s = src[i]; t = dst[i]; } else { s = t = i - Ee_; }
  float w = __expf(glog[i] - gmax[t]);
  if (lane == 0) atomicAdd(&gsum[t], w);
  float4 gv = *(const float4*)(g_h + (size_t)s * Dd_ + lane * 4);
  float* gn = GN + (size_t)t * Dd_ + lane * 4;
  atomicAdd(gn + 0, w * gv.x);
  atomicAdd(gn + 1, w * gv.y);
  atomicAdd(gn + 2, w * gv.z);
  atomicAdd(gn + 3, w * gv.w);
}

__global__ void gat_node(const float* __restrict__ GN, const float* __restrict__ gsum,
                         float* __restrict__ xnb) {
  int n = blockIdx.x, d = threadIdx.x;
  xnb[(size_t)n * Dd_ + d] += GN[(size_t)n * Dd_ + d] / (gsum[n] + 1e-16f) * 0.25f;
}

// ---------------------------------------------------------------------------
// Fusion: gated concat, then LN -> relu -> residual -> LN (wave per node)
// ---------------------------------------------------------------------------
__global__ void cat_kernel(const float* __restrict__ xs, const float* __restrict__ xa,
                           const float* __restrict__ xn,
                           const float* __restrict__ g_s, const float* __restrict__ g_a,
                           const float* __restrict__ g_n, float* __restrict__ cat) {
  size_t i = (size_t)blockIdx.x * blockDim.x + threadIdx.x;
  if (i >= (size_t)Nn_ * Dd_) return;
  size_t n = i >> 7, d = i & (Dd_ - 1);
  float gs = 1.f / (1.f + __expf(-g_s[0]));
  float ga = 1.f / (1.f + __expf(-g_a[0]));
  float gn = 1.f / (1.f + __expf(-g_n[0]));
  float* row = cat + n * (3 * Dd_);
  row[d]            = xs[i] * gs;
  row[Dd_ + d]      = xa[i] * ga;
  row[2 * Dd_ + d]  = xn[i] * gn;
}

__global__ void fuse_ln(const float* __restrict__ pre, const float* __restrict__ x,
                        const float* __restrict__ fg, const float* __restrict__ fb,
                        const float* __restrict__ ng, const float* __restrict__ nb,
                        float* __restrict__ out) {
  int wave = threadIdx.x >> 5, lane = threadIdx.x & 31;
  int n = blockIdx.x * 8 + wave;
  if (n >= Nn_) return;
  float4 p4 = *(const float4*)(pre + (size_t)n * Dd_ + lane * 4);
  float s1 = p4.x + p4.y + p4.z + p4.w;
  float s2 = p4.x * p4.x + p4.y * p4.y + p4.z * p4.z + p4.w * p4.w;
  s1 = waveSum(s1); s2 = waveSum(s2);
  float m = s1 * (1.f / Dd_);
  float inv = rsqrtf(s2 * (1.f / Dd_) - m * m + 1e-5f);
  float4 g4 = *(const float4*)(fg + lane * 4);
  float4 b4 = *(const float4*)(fb + lane * 4);
  float4 x4 = *(const float4*)(x + (size_t)n * Dd_ + lane * 4);
  float y0 = x4.x + fmaxf((p4.x - m) * inv * g4.x + b4.x, 0.f);
  float y1 = x4.y + fmaxf((p4.y - m) * inv * g4.y + b4.y, 0.f);
  float y2 = x4.z + fmaxf((p4.z - m) * inv * g4.z + b4.z, 0.f);
  float y3 = x4.w + fmaxf((p4.w - m) * inv * g4.w + b4.w, 0.f);
  float t1 = y0 + y1 + y2 + y3;
  float t2 = y0 * y0 + y1 * y1 + y2 * y2 + y3 * y3;
  t1 = waveSum(t1); t2 = waveSum(t2);
  float m2 = t1 * (1.f / Dd_);
  float inv2 = rsqrtf(t2 * (1.f / Dd_) - m2 * m2 + 1e-5f);
  float4 ng4 = *(const float4*)(ng + lane * 4);
  float4 nb4 = *(const float4*)(nb + lane * 4);
  float4 o;
  o.x = (y0 - m2) * inv2 * ng4.x + nb4.x;
  o.y = (y1 - m2) * inv2 * ng4.y + nb4.y;
  o.z = (y2 - m2) * inv2 * ng4.z + nb4.z;
  o.w = (y3 - m2) * inv2 * ng4.w + nb4.w;
  *(float4*)(out + (size_t)n * Dd_ + lane * 4) = o;
}

// ---------------------------------------------------------------------------
// Host side
// ---------------------------------------------------------------------------
static inline void fillf(float* p, long long n, float v, hipStream_t s) {
  int blocks = (int)((n + 255) / 256);
  if (blocks > 4096) blocks = 4096;
  fill_kernel<<<blocks, 256, 0, s>>>(p, n, v);
}

static inline void launch_gemm(const float* A, int lda, const float* Wm, int ldw, int wcol0,
                               const float* bias, float* C, int ldc, int M, int K, int Nout,
                               int acc, int relu, hipStream_t s) {
  dim3 grid(Nout / 16, (M + 127) / 128);
  gemm_wmma<<<grid, 256, 0, s>>>(A, lda, Wm, ldw, wcol0, bias, C, ldc, M, K, Nout, acc, relu);
}

extern "C" void kernel_launch(void* const* d_in, const int* in_sizes, int n_in,
                              void* d_out, int out_size, void* d_ws, size_t ws_size,
                              hipStream_t stream) {
  const float* x       = (const float*)d_in[0];
  const int*   ei      = (const int*)d_in[1];
  const float* ea      = (const float*)d_in[2];
  const float* sage_Wl = (const float*)d_in[3];
  const float* sage_Wr = (const float*)d_in[4];
  const float* sage_b  = (const float*)d_in[5];
  const float* tq_W    = (const float*)d_in[6];
  const float* tq_b    = (const float*)d_in[7];
  const float* tk_W    = (const float*)d_in[8];
  const float* tk_b    = (const float*)d_in[9];
  const float* tv_W    = (const float*)d_in[10];
  const float* tv_b    = (const float*)d_in[11];
  const float* te_W    = (const float*)d_in[12];
  const float* tskip_W = (const float*)d_in[13];
  const float* tskip_b = (const float*)d_in[14];
  const float* gat_W   = (const float*)d_in[15];
  const float* att_src = (const float*)d_in[16];
  const float* att_dst = (const float*)d_in[17];
  const float* gat_bias= (const float*)d_in[18];
  const float* g_short = (const float*)d_in[19];
  const float* g_attn  = (const float*)d_in[20];
  const float* g_neigh = (const float*)d_in[21];
  const float* fus_W   = (const float*)d_in[22];
  const float* fus_b   = (const float*)d_in[23];
  const float* fus_g   = (const float*)d_in[24];
  const float* fus_beta= (const float*)d_in[25];
  const float* norm_g  = (const float*)d_in[26];
  const float* norm_b  = (const float*)d_in[27];
  float* out = (float*)d_out;

  const int* src = ei;
  const int* dst = ei + Ee_;

  const size_t ND = (size_t)Nn_ * Dd_;
  float* W = (float*)d_ws;
  float* x_short = W;
  float* x_attn  = W + ND;
  float* x_nb    = W + 2 * ND;
  float* S       = W + 3 * ND;          // stage scratch (reused)

  // ---------------- Stage A: SAGE ----------------
  {
    float* agg = S;
    float* cnt = S + ND;
    fillf(agg, (long long)ND, 0.f, stream);
    fillf(cnt, Nn_, 0.f, stream);
    sage_scatter<<<Ee_, Dd_, 0, stream>>>(src, dst, x, agg, cnt);
    mean_kernel<<<(int)((ND + 255) / 256), 256, 0, stream>>>(agg, cnt);
    launch_gemm(agg, Dd_, sage_Wl, Dd_, 0, nullptr, x_short, Dd_, Nn_, Dd_, Dd_, 0, 0, stream);
    launch_gemm(x,   Dd_, sage_Wr, Dd_, 0, sage_b,  x_short, Dd_, Nn_, Dd_, Dd_, 1, 1, stream);
  }

  // ---------------- Stage B: TransformerConv ----------------
  {
    float* q_h  = S;
    float* k_h  = S + ND;
    float* v_h  = S + 2 * ND;
    float* VN   = S + 3 * ND;
    float* skip = S + 4 * ND;
    float* P    = S + 5 * ND;
    float* WA   = P + (size_t)Nn_ * ED_;
    float* tmax = WA + (size_t)Nn_ * ED_;
    float* tsum = tmax + Nn_;
    float* tlog = tsum + Nn_;

    launch_gemm(x, Dd_, tskip_W, Dd_, 0, tskip_b, skip, Dd_, Nn_, Dd_, Dd_, 0, 0, stream);
    fillf(x_attn, (long long)ND, 0.f, stream);

    for (int h = 0; h < Hh_; ++h) {
      launch_gemm(x, Dd_, tq_W, HD_, h * Dd_, tq_b + h * Dd_, q_h, Dd_, Nn_, Dd_, Dd_, 0, 0, stream);
      launch_gemm(x, Dd_, tk_W, HD_, h * Dd_, tk_b + h * Dd_, k_h, Dd_, Nn_, Dd_, Dd_, 0, 0, stream);
      launch_gemm(x, Dd_, tv_W, HD_, h * Dd_, tv_b + h * Dd_, v_h, Dd_, Nn_, Dd_, Dd_, 0, 0, stream);
      fillf(VN, (long long)ND, 0.f, stream);
      fillf(WA, (long long)Nn_ * ED_, 0.f, stream);
      fillf(tmax, Nn_, -3.0e38f, stream);
      fillf(tsum, Nn_, 0.f, stream);
      trans_P<<<(Nn_ + 7) / 8, 256, 0, stream>>>(q_h, te_W, h, P);
      trans_logits<<<(Ee_ + 7) / 8, 256, 0, stream>>>(src, dst, q_h, k_h, v_h, ea, P, tlog, tmax);
      trans_accum<<<(Ee_ + 7) / 8, 256, 0, stream>>>(src, dst, v_h, ea, tlog, tmax, tsum, VN, WA);
      trans_node<<<Nn_, Dd_, 0, stream>>>(VN, WA, tsum, te_W, h, x_attn);
    }
    add_relu<<<(int)((ND + 255) / 256), 256, 0, stream>>>(x_attn, skip, x_attn, (long long)ND);
  }

  // ---------------- Stage C: GAT ----------------
  {
    float* g_h  = S;
    float* GN   = S + ND;
    float* a_s  = S + 2 * ND;
    float* a_d  = a_s + Nn_;
    float* gmax = a_d + Nn_;
    float* gsum = gmax + Nn_;
    float* glog = gsum + Nn_;

    fillf(x_nb, (long long)ND, 0.f, stream);
    for (int h = 0; h < Hh_; ++h) {
      launch_gemm(x, Dd_, gat_W, HD_, h * Dd_, nullptr, g_h, Dd_, Nn_, Dd_, Dd_, 0, 0, stream);
      fillf(GN, (long long)ND, 0.f, stream);
      fillf(gmax, Nn_, -3.0e38f, stream);
      fillf(gsum, Nn_, 0.f, stream);
      gat_ad<<<(Nn_ + 7) / 8, 256, 0, stream>>>(g_h, att_src, att_dst, h, a_s, a_d);
      gat_logits<<<(Ee_ + Nn_ + 255) / 256, 256, 0, stream>>>(src, dst, a_s, a_d, glog, gmax);
      gat_accum<<<(Ee_ + Nn_ + 7) / 8, 256, 0, stream>>>(src, dst, g_h, glog, gmax, gsum, GN);
      gat_node<<<Nn_, Dd_, 0, stream>>>(GN, gsum, x_nb);
    }
    bias_relu<<<(int)((ND + 255) / 256), 256, 0, stream>>>(x_nb, gat_bias, x_nb, (long long)ND);
  }

  // ---------------- Stage D: fusion + double LayerNorm ----------------
  {
    float* cat = S;                         // N x 384
    float* pre = S + (size_t)Nn_ * 3 * Dd_; // N x 128
    cat_kernel<<<(int)((ND + 255) / 256), 256, 0, stream>>>(
        x_short, x_attn, x_nb, g_short, g_attn, g_neigh, cat);
    launch_gemm(cat, 3 * Dd_, fus_W, Dd_, 0, fus_b, pre, Dd_, Nn_, 3 * Dd_, Dd_, 0, 0, stream);
    fuse_ln<<<(Nn_ + 7) / 8, 256, 0, stream>>>(pre, x, fus_g, fus_beta, norm_g, norm_b, out);
  }
}